// KMeansTokenizer_12704513261637
// MI455X (gfx1250) — compile-verified
//
#include <hip/hip_runtime.h>
#include <math.h>

typedef float v2f __attribute__((ext_vector_type(2)));
typedef float v8f __attribute__((ext_vector_type(8)));

#define NUM_CODES 128
#define CODE_DIM  64
#define BB        64
#define TT        8192
#define NTOK      (BB * TT)   // 524288

// LDS fragment buffer: [8 ntile][16 kstep][32 lane][2 floats] = 8192 floats = 32 KB
// Fragment-ready layout for V_WMMA_F32_16X16X4_F32 B operand:
//   lane holds N = (lane&15), K = 2*(lane>>4) + j   (j = 0,1 -> the two B VGPRs)

__global__ __launch_bounds__(256)
void vq_assign_kernel(const float* __restrict__ z, const float* __restrict__ cent,
                      float* __restrict__ out, unsigned int* __restrict__ counts) {
    __shared__ v2f  fragB[8 * 16 * 32];     // 32 KB
    __shared__ float cn2[NUM_CODES];
    __shared__ unsigned int hist[NUM_CODES];

    const int tid = threadIdx.x;

    // ---- stage centroids into fragment-ready LDS layout ----
    {
        float* fragF = (float*)fragB;
        for (int i = tid; i < 8192; i += 256) {
            const int j     = i & 1;
            const int lane  = (i >> 1) & 31;
            const int k     = (i >> 6) & 15;
            const int ntile = i >> 10;
            const int d = 4 * k + j + ((lane >> 4) << 1);
            const int n = ntile * 16 + (lane & 15);
            fragF[i] = cent[n * CODE_DIM + d];
        }
    }
    if (tid < NUM_CODES) {
        float s = 0.f;
        const float* c = cent + tid * CODE_DIM;
        #pragma unroll
        for (int d = 0; d < CODE_DIM; ++d) s += c[d] * c[d];
        cn2[tid]  = s;
        hist[tid] = 0u;
    }
    __syncthreads();

    const int lane    = tid & 31;
    const int wave    = tid >> 5;
    const int halfsel = lane >> 4;     // 0: M rows 0..7, 1: M rows 8..15 (for C layout)
    const int lm      = lane & 15;

    const int numPairs   = NTOK / 32;                 // 16384 (32 tokens per pair)
    const int waveId     = blockIdx.x * 8 + wave;
    const int waveStride = gridDim.x * 8;

    for (int pair = waveId; pair < numPairs; pair += waveStride) {
        const int b  = pair >> 8;                     // TT/32 = 256 pairs per batch row
        const int t0 = (pair & 255) << 5;             // first of 32 tokens
        const float* zb = z + (size_t)b * CODE_DIM * TT;

        // ---- load A fragments for two 16-token tiles (f32 16x4 layout) ----
        v2f a0[16], a1[16];
        #pragma unroll
        for (int k = 0; k < 16; ++k) {
            const int d0 = 4 * k + (halfsel << 1);
            const float* p0 = zb + (size_t)d0 * TT + t0 + lm;
            const float* p1 = zb + (size_t)(d0 + 1) * TT + t0 + lm;
            a0[k].x = p0[0];   a0[k].y = p1[0];
            a1[k].x = p0[16];  a1[k].y = p1[16];
        }

        float bestV0[8], bestV1[8];
        int   bestI0[8], bestI1[8];
        #pragma unroll
        for (int r = 0; r < 8; ++r) {
            bestV0[r] = 3.4e38f; bestV1[r] = 3.4e38f;
            bestI0[r] = 0;       bestI1[r] = 0;
        }

        for (int ntile = 0; ntile < 8; ++ntile) {
            v8f acc0 = {}; v8f acc1 = {};
            const v2f* fb = fragB + ntile * 16 * 32;
            #pragma unroll
            for (int k = 0; k < 16; ++k) {
                v2f bf = fb[k * 32 + lane];          // one ds_load_b64 per WMMA pair
                acc0 = __builtin_amdgcn_wmma_f32_16x16x4_f32(
                           false, a0[k], false, bf, (short)0, acc0, false, false);
                acc1 = __builtin_amdgcn_wmma_f32_16x16x4_f32(
                           false, a1[k], false, bf, (short)0, acc1, false, false);
            }
            const float c2 = cn2[ntile * 16 + lm];
            const int   n  = ntile * 16 + lm;
            #pragma unroll
            for (int r = 0; r < 8; ++r) {
                const float d0v = c2 - 2.0f * acc0[r];
                const float d1v = c2 - 2.0f * acc1[r];
                if (d0v < bestV0[r]) { bestV0[r] = d0v; bestI0[r] = n; }
                if (d1v < bestV1[r]) { bestV1[r] = d1v; bestI1[r] = n; }
            }
        }

        // ---- butterfly argmin across each 16-lane half; tie -> smaller index ----
        #pragma unroll
        for (int r = 0; r < 8; ++r) {
            float v0 = bestV0[r]; int i0 = bestI0[r];
            float v1 = bestV1[r]; int i1 = bestI1[r];
            #pragma unroll
            for (int m = 1; m <= 8; m <<= 1) {
                float ov0 = __shfl_xor(v0, m, 32); int oi0 = __shfl_xor(i0, m, 32);
                float ov1 = __shfl_xor(v1, m, 32); int oi1 = __shfl_xor(i1, m, 32);
                if (ov0 < v0 || (ov0 == v0 && oi0 < i0)) { v0 = ov0; i0 = oi0; }
                if (ov1 < v1 || (ov1 == v1 && oi1 < i1)) { v1 = ov1; i1 = oi1; }
            }
            // token (tile0) = t0 + r + 8*halfsel ; token (tile1) = +16
            if (lm == 0) {
                const int t = t0 + r + (halfsel << 3);
                out[(size_t)b * TT + t]      = (float)i0;
                out[(size_t)b * TT + t + 16] = (float)i1;
            }
            if (lm == r) {        // lanes r and 16+r: one add per token
                atomicAdd(&hist[i0], 1u);
                atomicAdd(&hist[i1], 1u);
            }
        }
    }

    __syncthreads();
    if (tid < NUM_CODES) atomicAdd(&counts[tid], hist[tid]);
}

__global__ __launch_bounds__(128)
void vq_stats_kernel(const unsigned int* __restrict__ counts, float* __restrict__ out) {
    __shared__ float sH[NUM_CODES];
    __shared__ float sU[NUM_CODES];
    const int tid = threadIdx.x;
    const float c = (float)counts[tid];
    const float p = fmaxf(c / (float)NTOK, 1e-12f);
    sH[tid] = -p * logf(p);
    sU[tid] = (c > 0.f) ? 1.f : 0.f;
    __syncthreads();
    for (int s = 64; s > 0; s >>= 1) {
        if (tid < s) { sH[tid] += sH[tid + s]; sU[tid] += sU[tid + s]; }
        __syncthreads();
    }
    if (tid == 0) {
        out[NTOK]     = expf(sH[0]);              // perplexity
        out[NTOK + 1] = sU[0] / (float)NUM_CODES; // usage
    }
}

extern "C" void kernel_launch(void* const* d_in, const int* in_sizes, int n_in,
                              void* d_out, int out_size, void* d_ws, size_t ws_size,
                              hipStream_t stream) {
    const float* z    = (const float*)d_in[0];   // [64, 64, 8192] f32
    const float* cent = (const float*)d_in[1];   // [128, 64] f32
    float* out = (float*)d_out;                  // 524288 idx + perp + usage
    unsigned int* counts = (unsigned int*)d_ws;  // 128 bins

    hipMemsetAsync(counts, 0, NUM_CODES * sizeof(unsigned int), stream);
    vq_assign_kernel<<<1024, 256, 0, stream>>>(z, cent, out, counts);
    vq_stats_kernel<<<1, 128, 0, stream>>>(counts, out);
}